// rcc_loss_77936476553574
// MI455X (gfx1250) — compile-verified
//
#include <hip/hip_runtime.h>
#include <hip/hip_bf16.h>

// CDNA5 / gfx1250.  Pure HBM-bound reduction (128 MiB single pass) + tiny finalize.
// Uses V_WMMA_F32_16X16X4_F32 (exact f32) as the wave-level cross-lane reduction.

typedef __attribute__((ext_vector_type(2))) float v2f;
typedef __attribute__((ext_vector_type(8))) float v8f;

// Reduce one f32 value across all 32 lanes of a wave using WMMA.
// A[m][k]: lane m supplies its partial in one K slot (other slot zero);
// B = all ones  =>  C[m][n] = p[m] + p[m+16] for every n.
// Per-lane sum of the 8 C VGPRs gives the half-wave totals; one xor-16
// shuffle-add yields the full 32-lane sum in every lane (bit-exact f32).
// Requires EXEC == all ones at the call site.
__device__ __forceinline__ float wave_reduce_f32_wmma(float v) {
    v2f a; a[0] = v;    a[1] = 0.0f;
    v2f b; b[0] = 1.0f; b[1] = 1.0f;
    v8f c = {};
    c = __builtin_amdgcn_wmma_f32_16x16x4_f32(
            /*neg_a=*/false, a, /*neg_b=*/false, b,
            /*c_mod=*/(short)0, c, /*reuse_a=*/false, /*reuse_b=*/false);
    float s = ((c[0] + c[1]) + (c[2] + c[3])) + ((c[4] + c[5]) + (c[6] + c[7]));
    s += __shfl_xor(s, 16, 32);
    return s;
}

// Pass 1: one block per row; coalesced float4 streaming; per-row sum -> d_ws.
__global__ void rcc_rowsum_kernel(const float* __restrict__ x,
                                  float* __restrict__ rowsum, int D) {
    const size_t row = (size_t)blockIdx.x;
    const float4* __restrict__ xr =
        reinterpret_cast<const float4*>(x + row * (size_t)D);
    const int nvec = D >> 2;

    float acc = 0.0f;
    for (int i = threadIdx.x; i < nvec; i += blockDim.x) {   // uniform trip count
        float4 v = xr[i];
        acc += (v.x + v.y) + (v.z + v.w);
    }

    // EXEC is all-ones here (no divergence yet) -> WMMA legal.
    float wsum = wave_reduce_f32_wmma(acc);

    __shared__ float sm[32];
    const int lane = threadIdx.x & 31;
    const int wave = threadIdx.x >> 5;
    if (lane == 0) sm[wave] = wsum;
    __syncthreads();
    if (threadIdx.x == 0) {
        float s = 0.0f;
        const int nw = (int)(blockDim.x >> 5);
        for (int w = 0; w < nw; ++w) s += sm[w];
        rowsum[row] = s;
    }
}

__device__ __forceinline__ float block_reduce(float v, float* red) {
    const int t = threadIdx.x;
    red[t] = v;
    __syncthreads();
    for (int s = (int)(blockDim.x >> 1); s > 0; s >>= 1) {
        if (t < s) red[t] += red[t + s];
        __syncthreads();
    }
    float r = red[0];
    __syncthreads();
    return r;
}

// Pass 2: single block. Masked/unmasked reductions over the 8192 row sums,
// then the logaddexp term over x[B-1] (4096 elems). Fixed-order -> deterministic.
__global__ void rcc_finalize_kernel(const float* __restrict__ x,
                                    const int* __restrict__ lbl,
                                    const float* __restrict__ rowsum,
                                    float* __restrict__ out, int B, int D) {
    __shared__ float red[1024];
    __shared__ float sh_avg_pos, sh_avg_neg, sh_npos;

    const int t  = threadIdx.x;
    const int nt = (int)blockDim.x;
    const int s_last = lbl[B - 1];

    float sum_all = 0.0f, pos_sum = 0.0f, npos = 0.0f;
    for (int i = t; i < B; i += nt) {
        float rs = rowsum[i];
        sum_all += rs;
        if (lbl[i] == s_last) { pos_sum += rs; npos += 1.0f; }
    }
    sum_all = block_reduce(sum_all, red);
    pos_sum = block_reduce(pos_sum, red);
    npos    = block_reduce(npos, red);

    if (t == 0) {
        const float n_pos = npos;
        const float n_neg = (float)B - npos;
        sh_avg_pos = pos_sum / (n_pos * (float)D);
        sh_avg_neg = (sum_all - pos_sum) / (n_neg * (float)D);
        sh_npos    = npos;
    }
    __syncthreads();

    const float avg_pos = sh_avg_pos;
    const float avg_neg = sh_avg_neg;
    const float n_pos   = sh_npos;
    const float n_neg   = (float)B - n_pos;

    const float* __restrict__ xl = x + (size_t)(B - 1) * (size_t)D;
    float L = 0.0f, Sx = 0.0f;
    for (int i = t; i < D; i += nt) {
        float xv = xl[i];
        float va = xv * avg_neg;
        float vb = xv * avg_pos;
        float mx = fmaxf(va, vb);
        float mn = fminf(va, vb);
        L  += mx + log1pf(expf(mn - mx));   // logaddexp(va, vb)
        Sx += xv;
    }
    L  = block_reduce(L, red);
    Sx = block_reduce(Sx, red);

    if (t == 0) {
        // loss.mean() = L/D - (n_neg*avg_neg + n_pos*avg_pos) * Sx / (D*B)
        float term = (n_neg * avg_neg + n_pos * avg_pos) * Sx
                   / ((float)D * (float)B);
        out[0] = L / (float)D - term;
    }
}

extern "C" void kernel_launch(void* const* d_in, const int* in_sizes, int n_in,
                              void* d_out, int out_size, void* d_ws, size_t ws_size,
                              hipStream_t stream) {
    const float* x   = (const float*)d_in[0];
    const int*   lbl = (const int*)d_in[1];
    const int B = in_sizes[1];
    const int D = in_sizes[0] / B;

    float* rowsum = (float*)d_ws;   // B floats of scratch (32 KB)

    rcc_rowsum_kernel<<<B, 256, 0, stream>>>(x, rowsum, D);
    rcc_finalize_kernel<<<1, 1024, 0, stream>>>(x, lbl, rowsum, (float*)d_out, B, D);
}